// EmbeddingLoss_20083267076316
// MI455X (gfx1250) — compile-verified
//
#include <hip/hip_runtime.h>
#include <hip/hip_bf16.h>

// Problem constants (from reference setup_inputs)
#define B_    2
#define F_    3
#define C_    16
#define H_    64
#define W_    64
#define N_    (H_ * W_)       // 4096 pixels per frame
#define NPAIR 3               // frame pairs: (0,1),(0,2),(1,2)

typedef __attribute__((ext_vector_type(2))) float v2f;
typedef __attribute__((ext_vector_type(8))) float v8f;

// ---------------------------------------------------------------------------
// Zero the per-(b,pair) accumulators (graph-safe, deterministic each call).
// ---------------------------------------------------------------------------
__global__ void EmbLoss_init_kernel(float* __restrict__ accum) {
    if (threadIdx.x < 2 * B_ * NPAIR) accum[threadIdx.x] = 0.0f;
}

// ---------------------------------------------------------------------------
// Transpose (b,f,c,n) -> (b,f,n,c) so a pixel's 16 channels are contiguous
// (64B rows -> aligned b64 fragment loads); precompute squared norms and
// mask-folded ids (vid = mask ? id : 0; validity test becomes vid > 0).
// ---------------------------------------------------------------------------
__global__ void EmbLoss_prep_kernel(const float* __restrict__ emb,
                                    const int* __restrict__ ids,
                                    const unsigned char* __restrict__ msk,
                                    float* __restrict__ trans,
                                    float* __restrict__ sq,
                                    int* __restrict__ vid) {
    int idx = blockIdx.x * blockDim.x + threadIdx.x;   // over B_*F_*N_
    if (idx >= B_ * F_ * N_) return;
    int bf = idx / N_;
    int n  = idx - bf * N_;
    const float* src = emb + (size_t)bf * C_ * N_ + n;
    float* dst = trans + (size_t)idx * C_;
    float s = 0.0f;
    #pragma unroll
    for (int c = 0; c < C_; ++c) {
        float v = src[(size_t)c * N_];
        dst[c] = v;
        s += v * v;
    }
    sq[idx] = s;
    vid[idx] = msk[idx] ? ids[idx] : 0;
}

// ---------------------------------------------------------------------------
// Main fused kernel: one wave per 16-row strip of one (b, frame-pair) gram
// matrix. Gram tile via 4x V_WMMA_F32_16X16X4_F32 (K=16), loss fused in
// registers, wave32 shuffle reduction, atomic per-(b,pair) accumulation.
// ---------------------------------------------------------------------------
__global__ void __launch_bounds__(256)
EmbLoss_pair_kernel(const float* __restrict__ trans,
                    const float* __restrict__ sq,
                    const int* __restrict__ vid,
                    float* __restrict__ accum) {
    const int wave = threadIdx.x >> 5;
    const int lane = threadIdx.x & 31;
    const int hi   = lane >> 4;      // half-wave select (K / row-group split)
    const int lo   = lane & 15;      // matrix row/col within tile

    const int b    = blockIdx.z;
    const int pair = blockIdx.y;
    const int f1   = (pair < 2) ? 0 : 1;
    const int f2   = (pair == 0) ? 1 : 2;
    const int rowBase = (blockIdx.x * 8 + wave) * 16;

    const float* tA  = trans + (size_t)(b * F_ + f1) * N_ * C_;
    const float* tB  = trans + (size_t)(b * F_ + f2) * N_ * C_;
    const float* sq1 = sq  + (b * F_ + f1) * N_;
    const float* sq2 = sq  + (b * F_ + f2) * N_;
    const int*   id1 = vid + (b * F_ + f1) * N_;
    const int*   id2 = vid + (b * F_ + f2) * N_;

    // A fragments: 16x4 f32 per chunk q. Lane layout (ISA 7.12.2):
    //   lanes 0-15 hold row M=lo with K = 4q+{0,1}; lanes 16-31 K = 4q+{2,3}.
    v2f af[4];
    #pragma unroll
    for (int q = 0; q < 4; ++q)
        af[q] = *(const v2f*)(tA + (size_t)(rowBase + lo) * C_ + 4 * q + 2 * hi);

    // Row-side scalars for the 8 D elements this lane owns (M = r + 8*hi).
    float sq1v[8]; int id1v[8];
    #pragma unroll
    for (int r = 0; r < 8; ++r) {
        int row = rowBase + 8 * hi + r;
        sq1v[r] = sq1[row];
        id1v[r] = id1[row];
    }

    float sum = 0.0f, cnt = 0.0f;

    for (int ct = 0; ct < N_ / 16; ++ct) {
        const int col = ct * 16 + lo;          // D column owned by this lane
        const float sq2v = sq2[col];
        const int   id2v = id2[col];

        // Gram tile: acc[r](lane) = e1[rowBase+r+8*hi] . e2[col]
        v8f acc = {};
        #pragma unroll
        for (int q = 0; q < 4; ++q) {
            // B fragment 4x16: lanes 0-15 rows K=4q+{0,1}, lanes 16-31 K=4q+{2,3}
            v2f bf = *(const v2f*)(tB + (size_t)col * C_ + 4 * q + 2 * hi);
            acc = __builtin_amdgcn_wmma_f32_16x16x4_f32(
                /*neg_a=*/false, af[q], /*neg_b=*/false, bf,
                /*c_mod=*/(short)0, acc, /*reuse_a=*/false, /*reuse_b=*/false);
        }

        // Fused contrastive loss (MARGIN = POS_W = NEG_W = 1).
        #pragma unroll
        for (int r = 0; r < 8; ++r) {
            float dist  = sq1v[r] + sq2v - 2.0f * acc[r];
            float comb  = (id1v[r] == id2v) ? dist : fmaxf(1.0f - dist, 0.0f);
            bool  valid = (id1v[r] > 0) & (id2v > 0);
            sum += valid ? comb : 0.0f;
            cnt += valid ? 1.0f : 0.0f;
        }
    }

    // Wave32 reduction.
    #pragma unroll
    for (int off = 16; off >= 1; off >>= 1) {
        sum += __shfl_xor(sum, off, 32);
        cnt += __shfl_xor(cnt, off, 32);
    }
    if (lane == 0) {
        atomicAdd(&accum[(b * NPAIR + pair) * 2 + 0], sum);
        atomicAdd(&accum[(b * NPAIR + pair) * 2 + 1], cnt);
    }
}

// ---------------------------------------------------------------------------
// Final scalar: per-(b,pair) normalization, then average over non-empty pairs.
// ---------------------------------------------------------------------------
__global__ void EmbLoss_final_kernel(const float* __restrict__ accum,
                                     float* __restrict__ out) {
    if (threadIdx.x == 0 && blockIdx.x == 0) {
        float total = 0.0f, count = 0.0f;
        #pragma unroll
        for (int p = 0; p < B_ * NPAIR; ++p) {
            float s = accum[2 * p + 0];
            float c = accum[2 * p + 1];
            if (c > 0.0f) { total += s / c; count += 1.0f; }
        }
        out[0] = (count > 0.0f) ? (total / count) : 0.0f;
    }
}

extern "C" void kernel_launch(void* const* d_in, const int* in_sizes, int n_in,
                              void* d_out, int out_size, void* d_ws, size_t ws_size,
                              hipStream_t stream) {
    const float*         emb = (const float*)d_in[0];          // (B,F,C,H,W) f32
    const int*           ids = (const int*)d_in[1];            // (B,F,1,H,W) i32
    const unsigned char* msk = (const unsigned char*)d_in[2];  // (B,F,1,H,W) bool

    // Workspace layout (~1.8 MB total)
    float* trans = (float*)d_ws;                               // B*F*N*C f32
    float* sqv   = trans + (size_t)B_ * F_ * N_ * C_;          // B*F*N  f32
    int*   vid   = (int*)(sqv + (size_t)B_ * F_ * N_);         // B*F*N  i32
    float* accum = (float*)(vid + (size_t)B_ * F_ * N_);       // 12     f32

    EmbLoss_init_kernel<<<1, 32, 0, stream>>>(accum);

    const int prep_n = B_ * F_ * N_;
    EmbLoss_prep_kernel<<<(prep_n + 255) / 256, 256, 0, stream>>>(
        emb, ids, msk, trans, sqv, vid);

    dim3 grid(N_ / 16 / 8, NPAIR, B_);   // 32 x 3 x 2 blocks, 8 waves each
    EmbLoss_pair_kernel<<<grid, 256, 0, stream>>>(trans, sqv, vid, accum);

    EmbLoss_final_kernel<<<1, 32, 0, stream>>>(accum, (float*)d_out);
}